// ContextBaseTailAttention_42262478192938
// MI455X (gfx1250) — compile-verified
//
#include <hip/hip_runtime.h>
#include <math.h>

// Problem constants (match reference)
#define BB   32
#define SS   4096
#define HH   512
#define CHUNKS 32                       // S-chunks per batch for split softmax
#define ROWS_PER_CHUNK (SS / CHUNKS)    // 128
#define WAVES_PER_WG 8                  // 256 threads, wave32
#define ROWS_PER_WAVE (ROWS_PER_CHUNK / WAVES_PER_WG) // 16

typedef float v2f __attribute__((ext_vector_type(2)));
typedef float v4f __attribute__((ext_vector_type(4)));
typedef float v8f __attribute__((ext_vector_type(8)));

// ---------------------------------------------------------------------------
// Kernel 1: q = h_t @ W  (32x512 = A[32,512] * B[512,512]) via fp32 WMMA.
// 64 waves, each computes one 16x16 tile of q, K-loop of 128 x K=4 steps.
// A-layout (16x4 f32): VGPR0: lanes0-15 K+0, lanes16-31 K+2; VGPR1: K+1/K+3.
// B-layout (4x16 f32): mirrored rows across the two VGPRs.
// C-layout: VGPR v -> row (v) lanes0-15, row (v+8) lanes16-31; N = lane&15.
// ---------------------------------------------------------------------------
__global__ __launch_bounds__(256) void qgemm_wmma_f32(
    const float* __restrict__ h_t,   // [32,512]
    const float* __restrict__ W,     // [512,512]
    float* __restrict__ q)           // [32,512]
{
  int wave = (blockIdx.x * 256 + threadIdx.x) >> 5;   // 0..63
  int lane = threadIdx.x & 31;
  int mt = wave >> 5;          // 0..1   (batch tile)
  int nt = wave & 31;          // 0..31  (output-col tile)
  int mrow = mt * 16;
  int ncol = nt * 16;
  int l16  = lane & 15;
  int hi   = lane >> 4;        // 0: lanes 0-15, 1: lanes 16-31

  v8f c = {};
  for (int k = 0; k < HH; k += 4) {
    v2f a, b;
    const float* ap = h_t + (size_t)(mrow + l16) * HH + (k + hi * 2);
    a.x = ap[0];               // A[m][k+2*hi]
    a.y = ap[1];               // A[m][k+2*hi+1]
    const float* bp = W + (size_t)(k + hi * 2) * HH + (ncol + l16);
    b.x = bp[0];               // B[k+2*hi][n]
    b.y = bp[HH];              // B[k+2*hi+1][n]
    c = __builtin_amdgcn_wmma_f32_16x16x4_f32(
        /*neg_a=*/false, a, /*neg_b=*/false, b,
        /*c_mod=*/(short)0, c, /*reuse_a=*/false, /*reuse_b=*/false);
  }
#pragma unroll
  for (int v = 0; v < 8; ++v) {
    int m = mrow + hi * 8 + v;
    q[(size_t)m * HH + ncol + l16] = c[v];
  }
}

// ---------------------------------------------------------------------------
// Kernel 2: single streaming pass over context:
//   - scores[b,s] = q[b] . context[b,s]  (wave-level dot + butterfly reduce)
//   - online softmax partials (m, l, acc[512]) per wave -> combined per WG
//   - new_buffer copy fused into the same read (h_t substituted at s==pos)
// Lane h-ownership: h = 4*lane + 128*j (+0..3), j=0..3 -> coalesced float4.
// ---------------------------------------------------------------------------
__global__ __launch_bounds__(256) void flash_pass(
    const float* __restrict__ ctx,     // [B,S,H]
    const float* __restrict__ h_t,     // [B,H]
    const float* __restrict__ q,       // [B,H] (ws)
    const int*   __restrict__ posp,    // scalar
    float* __restrict__ newbuf,        // [B,S,H] (d_out tail)
    float* __restrict__ part_acc,      // [B,CHUNKS,H]
    float* __restrict__ part_m,        // [B,CHUNKS]
    float* __restrict__ part_l)        // [B,CHUNKS]
{
  __shared__ float accLDS[WAVES_PER_WG][HH];   // 16 KB
  __shared__ float mLDS[WAVES_PER_WG];
  __shared__ float lLDS[WAVES_PER_WG];

  int b     = blockIdx.x / CHUNKS;
  int chunk = blockIdx.x % CHUNKS;
  int tid   = threadIdx.x;
  int wave  = tid >> 5;
  int lane  = tid & 31;
  int pos   = posp[0] % SS;

  // Preload this lane's slice of q[b] and h_t[b]
  v4f qv[4], hv[4];
  const v4f* q4  = (const v4f*)(q   + (size_t)b * HH);
  const v4f* ht4 = (const v4f*)(h_t + (size_t)b * HH);
#pragma unroll
  for (int j = 0; j < 4; ++j) { qv[j] = q4[lane + 32 * j]; hv[j] = ht4[lane + 32 * j]; }

  float m = -INFINITY, l = 0.f;
  v4f acc[4] = {};

  int s0 = chunk * ROWS_PER_CHUNK + wave * ROWS_PER_WAVE;
  for (int i = 0; i < ROWS_PER_WAVE; ++i) {
    int s = s0 + i;
    const v4f* row4 = (const v4f*)(ctx    + ((size_t)b * SS + s) * HH);
    v4f*       out4 = (v4f*)      (newbuf + ((size_t)b * SS + s) * HH);

    v4f r[4];
    float dot = 0.f;
#pragma unroll
    for (int j = 0; j < 4; ++j) {
      r[j] = __builtin_nontemporal_load(row4 + lane + 32 * j);
      dot += r[j].x * qv[j].x + r[j].y * qv[j].y + r[j].z * qv[j].z + r[j].w * qv[j].w;
    }
    // fused cyclic-buffer write (attention uses the ORIGINAL row)
    bool isPos = (s == pos);
#pragma unroll
    for (int j = 0; j < 4; ++j)
      __builtin_nontemporal_store(isPos ? hv[j] : r[j], out4 + lane + 32 * j);

    // wave32 butterfly reduction -> every lane holds the full dot
#pragma unroll
    for (int off = 16; off; off >>= 1) dot += __shfl_xor(dot, off, 32);

    // online softmax update
    float mnew  = fmaxf(m, dot);
    float scale = __expf(m - mnew);     // 0 on first row (m = -inf)
    float p     = __expf(dot - mnew);
    l = l * scale + p;
#pragma unroll
    for (int j = 0; j < 4; ++j) {
      acc[j].x = acc[j].x * scale + p * r[j].x;
      acc[j].y = acc[j].y * scale + p * r[j].y;
      acc[j].z = acc[j].z * scale + p * r[j].z;
      acc[j].w = acc[j].w * scale + p * r[j].w;
    }
    m = mnew;
  }

  // wave partials -> LDS
  v4f* aL = (v4f*)&accLDS[wave][0];
#pragma unroll
  for (int j = 0; j < 4; ++j) aL[lane + 32 * j] = acc[j];
  if (lane == 0) { mLDS[wave] = m; lLDS[wave] = l; }
  __syncthreads();

  // combine the 8 wave partials into one chunk partial (fixed order)
  float M = mLDS[0];
#pragma unroll
  for (int w = 1; w < WAVES_PER_WG; ++w) M = fmaxf(M, mLDS[w]);
  float sc[WAVES_PER_WG];
  float L = 0.f;
#pragma unroll
  for (int w = 0; w < WAVES_PER_WG; ++w) { sc[w] = __expf(mLDS[w] - M); L += lLDS[w] * sc[w]; }

  size_t pbase = ((size_t)b * CHUNKS + chunk) * HH;
  for (int h = tid; h < HH; h += 256) {
    float a = 0.f;
#pragma unroll
    for (int w = 0; w < WAVES_PER_WG; ++w) a += accLDS[w][h] * sc[w];
    part_acc[pbase + h] = a;
  }
  if (tid == 0) {
    part_m[(size_t)b * CHUNKS + chunk] = M;
    part_l[(size_t)b * CHUNKS + chunk] = L;
  }
}

// ---------------------------------------------------------------------------
// Kernel 3: merge chunk partials -> cntx[b,:]  (tiny: 32 WGs x 256 threads)
// ---------------------------------------------------------------------------
__global__ __launch_bounds__(256) void combine_pass(
    const float* __restrict__ part_acc,
    const float* __restrict__ part_m,
    const float* __restrict__ part_l,
    float* __restrict__ cntx)          // [B,H] (d_out head)
{
  int b = blockIdx.x;
  int tid = threadIdx.x;

  float M = -INFINITY;
  for (int c = 0; c < CHUNKS; ++c) M = fmaxf(M, part_m[(size_t)b * CHUNKS + c]);
  float L = 0.f;
  for (int c = 0; c < CHUNKS; ++c)
    L += part_l[(size_t)b * CHUNKS + c] * __expf(part_m[(size_t)b * CHUNKS + c] - M);
  float inv = 1.f / L;

  for (int h = tid; h < HH; h += 256) {
    float a = 0.f;
    for (int c = 0; c < CHUNKS; ++c)
      a += part_acc[((size_t)b * CHUNKS + c) * HH + h] *
           __expf(part_m[(size_t)b * CHUNKS + c] - M);
    cntx[(size_t)b * HH + h] = a * inv;
  }
}

// ---------------------------------------------------------------------------
extern "C" void kernel_launch(void* const* d_in, const int* in_sizes, int n_in,
                              void* d_out, int out_size, void* d_ws, size_t ws_size,
                              hipStream_t stream) {
  (void)in_sizes; (void)n_in; (void)out_size; (void)ws_size;
  const float* h_t = (const float*)d_in[0];  // [32,512]
  const float* ctx = (const float*)d_in[1];  // [32,4096,512]
  const float* W   = (const float*)d_in[2];  // [512,512]
  const int*   pos = (const int*)d_in[3];    // scalar

  float* out    = (float*)d_out;
  float* cntx   = out;                               // [32,512]
  float* newbuf = out + (size_t)BB * HH;             // [32,4096,512]

  float* w        = (float*)d_ws;
  float* q        = w;                               // 32*512
  float* part_acc = q + (size_t)BB * HH;             // 32*CHUNKS*512
  float* part_m   = part_acc + (size_t)BB * CHUNKS * HH;
  float* part_l   = part_m + (size_t)BB * CHUNKS;

  // 1) q = h_t @ W via fp32 WMMA (64 waves)
  qgemm_wmma_f32<<<8, 256, 0, stream>>>(h_t, W, q);
  // 2) single streaming pass: scores + online softmax + fused buffer copy
  flash_pass<<<BB * CHUNKS, 256, 0, stream>>>(ctx, h_t, q, pos,
                                              newbuf, part_acc, part_m, part_l);
  // 3) merge split-softmax partials -> cntx
  combine_pass<<<BB, 256, 0, stream>>>(part_acc, part_m, part_l, cntx);
}